// VectorQuantizer_70806830842300
// MI455X (gfx1250) — compile-verified
//
#include <hip/hip_runtime.h>

typedef __attribute__((ext_vector_type(2))) float v2f;
typedef __attribute__((ext_vector_type(8))) float v8f;

#define K_CODES 2048
#define D_DIM   512
#define ROWS_PER_WG 128
#define WAVES_PER_WG 8
#define EPAD 548            // emb LDS row stride (548 % 64 == 36 -> conflict-free A b64 reads)
#define ZT   (D_DIM * 16)   // floats per z tile (one wave's 16 rows)
#define NTOT (16384 * 512)  // total elements for the mean

// ---------------- kernel 1: e_sq[k] = ||emb_k||^2 ----------------
__global__ void vq_esq(const float* __restrict__ emb, float* __restrict__ esq) {
    int k = blockIdx.x * blockDim.x + threadIdx.x;
    if (k < K_CODES) {
        const float* row = emb + (size_t)k * D_DIM;
        float s = 0.f;
#pragma unroll 8
        for (int d = 0; d < D_DIM; ++d) { float v = row[d]; s = fmaf(v, v, s); }
        esq[k] = s;
    }
}

// ---------------- kernel 2: main VQ (WMMA fp32) ----------------
__launch_bounds__(256, 1)
__global__ void vq_main(const float* __restrict__ z, const float* __restrict__ emb,
                        const float* __restrict__ esq_g, float* __restrict__ out,
                        double* __restrict__ partials) {
    __shared__ float z_s[WAVES_PER_WG * ZT];  // 256 KB: 8 tiles, layout [tile][d][r16]
    __shared__ float emb_s[16 * EPAD];        // ~35 KB: [code][d] padded
    __shared__ float esq_s[16];

    const int tid = threadIdx.x;
    const int w   = tid >> 5;        // wave id (0..7)
    const int l   = tid & 31;        // lane (wave32)
    const int hi  = l >> 4;          // half-wave
    const int lo  = l & 15;

    const int n0  = blockIdx.x * ROWS_PER_WG;   // 128 rows, same batch (128 | 1024)
    const int b   = n0 >> 10;
    const int hw0 = n0 & 1023;
    const float* zb = z + (size_t)b * 524288 + hw0;   // z[b][*][hw0..]

    // ---- stage z: rows r4..r4+3, strided d ----
    {
        const int r4  = (tid & 31) * 4;         // 0..124
        const int d0s = tid >> 5;               // 0..7
        float* dst0 = &z_s[(r4 >> 4) * ZT + (r4 & 15)];
        for (int d = d0s; d < D_DIM; d += 8) {
            float4 v = *(const float4*)(zb + (size_t)d * 1024 + r4);
            *(float4*)(dst0 + d * 16) = v;
        }
    }
    __syncthreads();

    float bestv = 3.4e38f;
    int   bestk = 0;

    const float* As = &emb_s[lo * EPAD + 2 * hi];           // A: code=lo, d=d0+2*hi{,+1}
    const float* Bs = &z_s[w * ZT + 2 * hi * 16 + lo];      // B: row=lo,  d=d0+2*hi{,+1}

    for (int k0 = 0; k0 < K_CODES; k0 += 16) {
        // ---- stage 16 codebook rows (shared by all 8 waves) ----
        {
            const int c  = tid >> 4;     // 0..15
            const int dq = tid & 15;     // 0..15
            const float* src = emb + (size_t)(k0 + c) * D_DIM;
            float* dst = &emb_s[c * EPAD];
#pragma unroll
            for (int i = 0; i < 8; ++i) {
                int d = 4 * dq + 64 * i;
                *(float4*)(dst + d) = *(const float4*)(src + d);
            }
            if (tid < 16) esq_s[tid] = esq_g[k0 + tid];
        }
        __syncthreads();

        // ---- 16x16 tile: dots(code m, row n) over D=512, fp32 WMMA ----
        // Fully unrolled: all DS addresses are base + immediate offset
        // (A: 16*i bytes <= 2032; B: 64*i floats <= 32 KB, both fit DS imm16).
        v8f acc = {0.f, 0.f, 0.f, 0.f, 0.f, 0.f, 0.f, 0.f};
#pragma unroll
        for (int i = 0; i < 128; ++i) {
            v2f a = *(const v2f*)(As + 4 * i);              // ds_load_b64, imm offset
            v2f bf;
            bf.x = Bs[64 * i];                              // ds_load_b32, imm offset
            bf.y = Bs[64 * i + 16];                         // ds_load_b32, imm offset
            acc = __builtin_amdgcn_wmma_f32_16x16x4_f32(
                false, a, false, bf, (short)0, acc, false, false);
        }

        // ---- fold e_sq - 2*dot, running argmin (lane holds m = j + 8*hi, n = lo) ----
#pragma unroll
        for (int j = 0; j < 8; ++j) {
            float sc = fmaf(-2.0f, acc[j], esq_s[j + 8 * hi]);
            int kk = k0 + j + 8 * hi;
            if (sc < bestv) { bestv = sc; bestk = kk; }
        }
        __syncthreads();   // before emb_s is overwritten
    }

    // ---- merge the two half-wave candidates for column n = lo (tie -> lower k) ----
    {
        float ov = __shfl_xor(bestv, 16);
        int   ok = __shfl_xor(bestk, 16);
        if (ov < bestv || (ov == bestv && ok < bestk)) { bestv = ov; bestk = ok; }
    }
    const int myidx = __shfl(bestk, lo);   // winning code for this lane's output row r = lo

    // ---- gather emb[idx], write NCHW output, accumulate SSE in fp64 ----
    double sse = 0.0;
    const float* qrow = emb + (size_t)myidx * D_DIM;
    float* outp = out + (size_t)b * 524288 + hw0 + w * 16 + lo;
    const float* zls = &z_s[w * ZT + lo];
#pragma unroll 4
    for (int m = 0; m < 256; ++m) {
        int d = hi + 2 * m;                 // lanes cover 16 rows x both d-parities
        float q  = qrow[d];
        float zz = zls[d * 16];
        float df = q - zz;
        sse += (double)df * (double)df;
        outp[(size_t)d * 1024] = q;         // 16 consecutive floats per d across lanes
    }

    // ---- deterministic per-wave partial ----
#pragma unroll
    for (int off = 16; off > 0; off >>= 1) sse += __shfl_xor(sse, off);
    if (l == 0) partials[blockIdx.x * WAVES_PER_WG + w] = sse;
}

// ---------------- kernel 3: reduce 1024 partials -> vq_loss ----------------
__global__ void vq_finalize(const double* __restrict__ partials, float* __restrict__ loss_out) {
    __shared__ double red[256];
    int t = threadIdx.x;
    red[t] = partials[t] + partials[t + 256] + partials[t + 512] + partials[t + 768];
    __syncthreads();
    for (int off = 128; off > 0; off >>= 1) {
        if (t < off) red[t] += red[t + off];
        __syncthreads();
    }
    if (t == 0) loss_out[0] = (float)(red[0] * (1.25 / (double)NTOT));
}

extern "C" void kernel_launch(void* const* d_in, const int* in_sizes, int n_in,
                              void* d_out, int out_size, void* d_ws, size_t ws_size,
                              hipStream_t stream) {
    const float* z   = (const float*)d_in[0];   // [16,512,32,32]
    const float* emb = (const float*)d_in[1];   // [2048,512]
    float* out = (float*)d_out;                 // 8388608 quantized + 1 loss

    double* partials = (double*)d_ws;                    // 1024 doubles (8 KB)
    float*  esq      = (float*)((char*)d_ws + 8192);     // 2048 floats  (8 KB)

    vq_esq<<<8, 256, 0, stream>>>(emb, esq);
    vq_main<<<16384 / ROWS_PER_WG, 256, 0, stream>>>(z, emb, esq, out, partials);
    vq_finalize<<<1, 256, 0, stream>>>(partials, out + 8388608);
}